// Net_31095563223317
// MI455X (gfx1250) — compile-verified
//
#include <hip/hip_runtime.h>
#include <hip/hip_bf16.h>

typedef __attribute__((ext_vector_type(2))) float v2f;
typedef __attribute__((ext_vector_type(8))) float v8f;

#define D_FEAT 512
#define HIDDEN 16
#define NCLS   40

// ---------------------------------------------------------------------------
// degree / norm helpers
// ---------------------------------------------------------------------------
__global__ void init_deg_kernel(float* __restrict__ deg, int n) {
    int i = blockIdx.x * blockDim.x + threadIdx.x;
    if (i < n) deg[i] = 1.0f;                       // self-loop weight
}

__global__ void scatter_deg_kernel(const int* __restrict__ col,
                                   const float* __restrict__ w,
                                   float* __restrict__ deg, int E) {
    int e = blockIdx.x * blockDim.x + threadIdx.x;
    if (e < E) atomicAdd(&deg[col[e]], w[e]);
}

__global__ void rsqrt_deg_kernel(float* __restrict__ deg, int n) {
    int i = blockIdx.x * blockDim.x + threadIdx.x;
    if (i < n) {
        float d = deg[i];
        deg[i] = (d > 0.0f) ? rsqrtf(d) : 0.0f;     // deg -> d^-1/2 in place
    }
}

__global__ void zero_kernel(float* __restrict__ p, int n) {
    int i = blockIdx.x * blockDim.x + threadIdx.x;
    if (i < n) p[i] = 0.0f;
}

// out[i*NCLS + j] = b2[j]
__global__ void init_out_kernel(float* __restrict__ out,
                                const float* __restrict__ b2, long n) {
    long i = (long)blockIdx.x * blockDim.x + threadIdx.x;
    if (i < n) out[i] = b2[i % NCLS];
}

// h2 = relu(agg1 + b1)
__global__ void relu_bias_kernel(const float* __restrict__ agg,
                                 const float* __restrict__ b1,
                                 float* __restrict__ h2, long n) {
    long i = (long)blockIdx.x * blockDim.x + threadIdx.x;
    if (i < n) {
        float v = agg[i] + b1[i % HIDDEN];
        h2[i] = v > 0.0f ? v : 0.0f;
    }
}

// ---------------------------------------------------------------------------
// GEMM1: h1[N,16] = x[N,512] @ W1[512,16]   (fp32 WMMA 16x16x4)
//   block = 128 threads = 4 waves, 64 rows per block
// ---------------------------------------------------------------------------
#define G1_TM 64
#define G1_KT 64
#define G1_LDA 68   // padded LDS row stride (floats) -> conflict-free frags

__global__ __launch_bounds__(128)
void gemm1_wmma_kernel(const float* __restrict__ x,
                       const float* __restrict__ W1,
                       float* __restrict__ h1, int nrows) {
    __shared__ float As[G1_TM * G1_LDA];
    __shared__ float Bs[D_FEAT * HIDDEN];           // 32 KB, loaded once

    const int tid  = threadIdx.x;
    const int lane = tid & 31;
    const int wave = tid >> 5;                      // 0..3
    const int half = lane >> 4;                     // 0/1 (lane group)
    const int lr   = lane & 15;                     // 0..15
    const int mbase = blockIdx.x * G1_TM;

    for (int i = tid; i < D_FEAT * HIDDEN; i += 128) Bs[i] = W1[i];

    v8f c = {};
    for (int k0 = 0; k0 < D_FEAT; k0 += G1_KT) {
        __syncthreads();                            // protect As reuse
        // cooperative float4 load of 64x64 A tile (row-clamped)
        for (int i = tid; i < G1_TM * G1_KT / 4; i += 128) {
            int r  = i / (G1_KT / 4);
            int cq = (i - r * (G1_KT / 4)) * 4;
            int gr = mbase + r;
            if (gr >= nrows) gr = nrows - 1;
            const float4 v =
                *reinterpret_cast<const float4*>(&x[(long)gr * D_FEAT + k0 + cq]);
            As[r * G1_LDA + cq + 0] = v.x;
            As[r * G1_LDA + cq + 1] = v.y;
            As[r * G1_LDA + cq + 2] = v.z;
            As[r * G1_LDA + cq + 3] = v.w;
            if (k0 + G1_KT < D_FEAT)                // global_prefetch_b8
                __builtin_prefetch(&x[(long)gr * D_FEAT + k0 + G1_KT + cq], 0, 1);
        }
        __syncthreads();

#pragma unroll
        for (int kk = 0; kk < G1_KT; kk += 4) {
            // A frag: lane holds A[M=lr][K = kk + 2*half + v]
            const int arow = wave * 16 + lr;
            v2f a;
            a.x = As[arow * G1_LDA + kk + 2 * half + 0];
            a.y = As[arow * G1_LDA + kk + 2 * half + 1];
            // B frag: lane holds B[K = kk + 2*half + v][N=lr]
            v2f b;
            b.x = Bs[(k0 + kk + 2 * half + 0) * HIDDEN + lr];
            b.y = Bs[(k0 + kk + 2 * half + 1) * HIDDEN + lr];
            c = __builtin_amdgcn_wmma_f32_16x16x4_f32(
                    false, a, false, b, (short)0, c, false, false);
        }
    }

    // C/D layout: VGPR v, lane L -> (M = v + 8*half, N = lr)
    const int mtile = mbase + wave * 16;
#pragma unroll
    for (int v = 0; v < 8; ++v) {
        int r = mtile + v + 8 * half;
        if (r < nrows) h1[(long)r * HIDDEN + lr] = c[v];
    }
}

// ---------------------------------------------------------------------------
// GEMM2: g2[N,40] = h2[N,16] @ W2[16,40]  (K=16, N padded 40->48, 3 n-tiles)
// ---------------------------------------------------------------------------
__global__ __launch_bounds__(128)
void gemm2_wmma_kernel(const float* __restrict__ h2,
                       const float* __restrict__ W2,
                       float* __restrict__ g2, int nrows) {
    const int tid  = threadIdx.x;
    const int lane = tid & 31;
    const int wave = tid >> 5;
    const int half = lane >> 4;
    const int lr   = lane & 15;
    const int mtile = (blockIdx.x * 4 + wave) * 16;  // wave-uniform
    if (mtile >= nrows) return;                      // whole wave exits

    v8f c0 = {}, c1 = {}, c2 = {};
#pragma unroll
    for (int kk = 0; kk < HIDDEN; kk += 4) {
        int r = mtile + lr;
        if (r >= nrows) r = nrows - 1;               // clamp, keep EXEC full
        const int kc = kk + 2 * half;
        v2f a;
        a.x = h2[(long)r * HIDDEN + kc + 0];
        a.y = h2[(long)r * HIDDEN + kc + 1];
        v2f b0, b1, b2;
        b0.x = W2[(kc + 0) * NCLS + lr];
        b0.y = W2[(kc + 1) * NCLS + lr];
        b1.x = W2[(kc + 0) * NCLS + 16 + lr];
        b1.y = W2[(kc + 1) * NCLS + 16 + lr];
        b2.x = (32 + lr < NCLS) ? W2[(kc + 0) * NCLS + 32 + lr] : 0.0f;
        b2.y = (32 + lr < NCLS) ? W2[(kc + 1) * NCLS + 32 + lr] : 0.0f;
        c0 = __builtin_amdgcn_wmma_f32_16x16x4_f32(false, a, false, b0, (short)0, c0, false, false);
        c1 = __builtin_amdgcn_wmma_f32_16x16x4_f32(false, a, false, b1, (short)0, c1, false, false);
        c2 = __builtin_amdgcn_wmma_f32_16x16x4_f32(false, a, false, b2, (short)0, c2, false, false);
    }

#pragma unroll
    for (int v = 0; v < 8; ++v) {
        int r = mtile + v + 8 * half;
        if (r < nrows) {
            g2[(long)r * NCLS + lr]      = c0[v];
            g2[(long)r * NCLS + 16 + lr] = c1[v];
            if (32 + lr < NCLS) g2[(long)r * NCLS + 32 + lr] = c2[v];
        }
    }
}

// ---------------------------------------------------------------------------
// message scatter: out[col] += dinv[row]*w*dinv[col] * h[row]
// virtual edges e >= E are self-loops (w = 1)
// ---------------------------------------------------------------------------
template <int F>
__global__ void msg_scatter_kernel(const int* __restrict__ row,
                                   const int* __restrict__ col,
                                   const float* __restrict__ w,
                                   const float* __restrict__ dinv,
                                   const float* __restrict__ h,
                                   float* __restrict__ out, int E, int Nn) {
    int gid = blockIdx.x * blockDim.x + threadIdx.x;
    int total = (E + Nn) * F;
    if (gid >= total) return;
    int e = gid / F;
    int j = gid - e * F;
    int r, c;
    float ww;
    if (e < E) {
        r = row[e];
        c = col[e];
        ww = w[e];
    } else {
        r = e - E;
        c = r;
        ww = 1.0f;
    }
    float nm = dinv[r] * ww * dinv[c];
    atomicAdd(out + (long)c * F + j, nm * h[(long)r * F + j]);
}

// ---------------------------------------------------------------------------
// launch
// ---------------------------------------------------------------------------
extern "C" void kernel_launch(void* const* d_in, const int* in_sizes, int n_in,
                              void* d_out, int out_size, void* d_ws, size_t ws_size,
                              hipStream_t stream) {
    const float* x    = (const float*)d_in[0];
    const int*   eidx = (const int*)d_in[1];
    const float* ew   = (const float*)d_in[2];
    const float* W1   = (const float*)d_in[3];
    const float* b1   = (const float*)d_in[4];
    const float* W2   = (const float*)d_in[5];
    const float* b2   = (const float*)d_in[6];
    float* out = (float*)d_out;

    const int N = in_sizes[0] / D_FEAT;   // 100000
    const int E = in_sizes[1] / 2;        // 3200000
    const int* row = eidx;
    const int* col = eidx + E;

    // workspace layout (floats): dinv[N] | h1[16N] | agg1[16N] | g2[40N]
    float* wsf  = (float*)d_ws;
    float* dinv = wsf;
    float* h1   = wsf + N;                // later reused as h2 (post-ReLU)
    float* agg1 = h1 + (long)N * HIDDEN;
    float* g2   = agg1 + (long)N * HIDDEN;

    const int T = 256;
    // 1) degrees (self-loop = 1.0) -> d^-1/2
    init_deg_kernel<<<(N + T - 1) / T, T, 0, stream>>>(dinv, N);
    scatter_deg_kernel<<<(E + T - 1) / T, T, 0, stream>>>(col, ew, dinv, E);
    rsqrt_deg_kernel<<<(N + T - 1) / T, T, 0, stream>>>(dinv, N);

    // 2) h1 = x @ W1  (fp32 WMMA)
    gemm1_wmma_kernel<<<(N + G1_TM - 1) / G1_TM, 128, 0, stream>>>(x, W1, h1, N);

    // 3) agg1 = scatter(norm * h1[row] -> col)
    long n16 = (long)N * HIDDEN;
    zero_kernel<<<(int)((n16 + T - 1) / T), T, 0, stream>>>(agg1, (int)n16);
    {
        int total = (E + N) * HIDDEN;
        msg_scatter_kernel<HIDDEN><<<(total + T - 1) / T, T, 0, stream>>>(
            row, col, ew, dinv, h1, agg1, E, N);
    }

    // 4) h2 = relu(agg1 + b1)   (written back into h1 buffer)
    relu_bias_kernel<<<(int)((n16 + T - 1) / T), T, 0, stream>>>(agg1, b1, h1, n16);

    // 5) g2 = h2 @ W2  (fp32 WMMA, N padded to 48)
    gemm2_wmma_kernel<<<(N + 63) / 64, 128, 0, stream>>>(h1, W2, g2, N);

    // 6) out = b2 + scatter(norm * g2[row] -> col)
    long n40 = (long)N * NCLS;
    init_out_kernel<<<(int)((n40 + T - 1) / T), T, 0, stream>>>(out, b2, n40);
    {
        int total = (E + N) * NCLS;
        msg_scatter_kernel<NCLS><<<(total + T - 1) / T, T, 0, stream>>>(
            row, col, ew, dinv, g2, out, E, N);
    }
}